// SE3PositionalEncoder_10814727651345
// MI455X (gfx1250) — compile-verified
//
#include <hip/hip_runtime.h>
#include <hip/hip_bf16.h>

// Problem constants (from reference)
#define BB 8
#define NN 4096
#define KK 16
#define NB 8
#define OUT_MUL 64
#define OUT_DIM (OUT_MUL * 3)      // 192
#define MAX_RADIUS 10.0f

// Tiling
#define TILE 16                    // row tile (WMMA M/N)
#define NUM_WAVES 8                // 256 threads / wave32
#define CHUNK_COLS 512             // columns of the score matrix staged in LDS at a time
#define TILES_PER_CHUNK (CHUNK_COLS / TILE)     // 32
#define TILES_PER_WAVE (TILES_PER_CHUNK / NUM_WAVES)  // 4
#define NUM_CHUNKS (NN / CHUNK_COLS)            // 8
#define STRIPES 16                 // selection threads per row
#define COLS_PER_STRIPE (CHUNK_COLS / STRIPES)  // 32
#define SCORE_STRIDE (CHUNK_COLS + 4)           // pad; 516*4B = 129*16B keeps f4 alignment

typedef float v2f __attribute__((ext_vector_type(2)));
typedef float v8f __attribute__((ext_vector_type(8)));

// ---------------------------------------------------------------------------
// Kernel 1: build augmented B-side columns: bcol[j] = (-2x, -2y, -2z, |p|^2)
// so that A_row(i)=(x,y,z,1) dot B_col(j) = -2 p_i.p_j + |p_j|^2  (row-constant
// |p_i|^2 omitted: it does not change per-row top-k ordering).
// ---------------------------------------------------------------------------
__global__ void se3_prep_bcol(const float* __restrict__ coords,
                              float4* __restrict__ bcol, int total) {
    int i = blockIdx.x * blockDim.x + threadIdx.x;
    if (i >= total) return;
    float x = coords[3 * i + 0];
    float y = coords[3 * i + 1];
    float z = coords[3 * i + 2];
    float4 v;
    v.x = -2.0f * x;
    v.y = -2.0f * y;
    v.z = -2.0f * z;
    v.w = x * x + y * y + z * z;
    bcol[i] = v;
}

// Insert (v, idx) into an ascending top-16 array (register-promoted).
__device__ __forceinline__ void insert_top16(float v, int idx,
                                             float* bv, int* bi) {
    if (v >= bv[15]) return;
    int p = 15;
    while (p > 0 && bv[p - 1] > v) {
        bv[p] = bv[p - 1];
        bi[p] = bi[p - 1];
        --p;
    }
    bv[p] = v;
    bi[p] = idx;
}

// ---------------------------------------------------------------------------
// Kernel 2: fused kNN (WMMA f32 16x16x4 scoring + hierarchical top-16 in LDS)
// + edge tensor product + mean aggregation. One block per (batch, 16 rows).
// ---------------------------------------------------------------------------
__global__ void __launch_bounds__(256)
se3_knn_tp_kernel(const float* __restrict__ coords,
                  const float4* __restrict__ bcol,
                  const float* __restrict__ tpW,   // [NB][OUT_MUL]
                  float* __restrict__ out) {       // [B][N][192]
    __shared__ float sW[NB * OUT_MUL];             // 2 KB
    __shared__ float rowC[TILE][3];                // row-tile coords
    __shared__ int   nbr[TILE][KK];                // selected neighbor indices
    __shared__ float G[TILE][NB * 3];              // per-row  sum_k rb_v * rhat_m
    // Score chunk (16 x 516 f32 = 33 KB); reused later for the candidate
    // merge buffers (16*16*16 vals + 16*16*16 idx = 32 KB <= 33 KB).
    __shared__ __align__(16) float scores[TILE * SCORE_STRIDE];

    const int tid    = threadIdx.x;
    const int lane   = tid & 31;
    const int waveId = tid >> 5;
    const int half   = lane >> 4;     // 0: lanes 0-15, 1: lanes 16-31
    const int lm     = lane & 15;

    const int b       = blockIdx.y;
    const int rowBase = blockIdx.x * TILE;

    // Stage W and the 16 row coords.
    for (int i = tid; i < NB * OUT_MUL; i += 256) sW[i] = tpW[i];
    if (tid < TILE) {
        const float* c = &coords[((size_t)b * NN + rowBase + tid) * 3];
        rowC[tid][0] = c[0];
        rowC[tid][1] = c[1];
        rowC[tid][2] = c[2];
    }
    __syncthreads();

    // A fragment (16x4 f32): lanes 0-15 hold K=0,1 (x,y); lanes 16-31 K=2,3 (z,1).
    v2f afrag;
    if (half == 0) {
        afrag.x = rowC[lm][0];
        afrag.y = rowC[lm][1];
    } else {
        afrag.x = rowC[lm][2];
        afrag.y = 1.0f;
    }

    // Private top-16 (value ascending) per (row, stripe) thread.
    const int selRow = tid >> 4;      // 0..15
    const int stripe = tid & 15;      // 0..15
    const int giSelf = rowBase + selRow;
    float bv[16];
    int   bi[16];
#pragma unroll
    for (int i = 0; i < 16; ++i) { bv[i] = INFINITY; bi[i] = -1; }

    const float4* bcolB = bcol + (size_t)b * NN;

    for (int chunk = 0; chunk < NUM_CHUNKS; ++chunk) {
        // ---- score CHUNK_COLS columns with WMMA f32 16x16x4 ----
        // Each wave owns exactly 4 tiles: batch the 4 B-fragment loads into
        // one clause, then issue 4 independent WMMAs back-to-back.
        {
            float2 bfr[TILES_PER_WAVE];
#pragma unroll
            for (int u = 0; u < TILES_PER_WAVE; ++u) {
                const int t   = waveId + u * NUM_WAVES;
                const int col = (chunk * TILES_PER_CHUNK + t) * TILE + lm;
                // lanes 0-15 need (-2x,-2y) = .xy; lanes 16-31 (-2z,|p|^2) = .zw
                const float2* p = reinterpret_cast<const float2*>(&bcolB[col]);
                bfr[u] = p[half];                       // single global_load_b64
            }
#pragma unroll
            for (int u = 0; u < TILES_PER_WAVE; ++u) {
                const int t = waveId + u * NUM_WAVES;
                v2f bfrag;
                bfrag.x = bfr[u].x;
                bfrag.y = bfr[u].y;
                v8f cacc = {};
                v8f d = __builtin_amdgcn_wmma_f32_16x16x4_f32(
                    false, afrag, false, bfrag, (short)0, cacc, false, false);
                // D layout: VGPR r -> row (r + 8*half), col lm within the tile.
#pragma unroll
                for (int r = 0; r < 8; ++r)
                    scores[(r + half * 8) * SCORE_STRIDE + t * TILE + lm] = d[r];
            }
        }
        // Prefetch next chunk's B-side columns into caches (global_prefetch_b8).
        if (chunk + 1 < NUM_CHUNKS)
            __builtin_prefetch(&bcolB[(chunk + 1) * CHUNK_COLS +
                                      (tid * 2 & (CHUNK_COLS - 1))], 0, 1);
        __syncthreads();

        // ---- stripe scan: each (row, stripe) thread folds 32 columns ----
        // 16B-aligned contiguous run -> 8 x ds_load_b128.
        {
            const int cbase    = stripe * COLS_PER_STRIPE;
            const int gcol0    = chunk * CHUNK_COLS + cbase;
            const float4* rowp = reinterpret_cast<const float4*>(
                &scores[selRow * SCORE_STRIDE + cbase]);
            for (int c4 = 0; c4 < COLS_PER_STRIPE / 4; ++c4) {
                float4 v4 = rowp[c4];
                const int g = gcol0 + c4 * 4;
                if (g + 0 != giSelf) insert_top16(v4.x, g + 0, bv, bi);
                if (g + 1 != giSelf) insert_top16(v4.y, g + 1, bv, bi);
                if (g + 2 != giSelf) insert_top16(v4.z, g + 2, bv, bi);
                if (g + 3 != giSelf) insert_top16(v4.w, g + 3, bv, bi);
            }
        }
        __syncthreads();                              // scores reused next chunk
    }

    // ---- dump per-stripe candidates into (reused) LDS ----
    float* candVal = scores;                                   // 4096 f32
    int*   candIdx = (int*)(scores + TILE * STRIPES * 16);     // 4096 i32
#pragma unroll
    for (int i = 0; i < 16; ++i) {
        candVal[(selRow * STRIPES + stripe) * 16 + i] = bv[i];
        candIdx[(selRow * STRIPES + stripe) * 16 + i] = bi[i];
    }
    __syncthreads();

    // ---- final per-row merge: 256 candidates -> 16 nearest ----
    if (tid < TILE) {
        float mv[16]; int mi[16];
#pragma unroll
        for (int i = 0; i < 16; ++i) { mv[i] = INFINITY; mi[i] = -1; }
        for (int j = 0; j < STRIPES * 16; ++j)
            insert_top16(candVal[tid * STRIPES * 16 + j],
                         candIdx[tid * STRIPES * 16 + j], mv, mi);
#pragma unroll
        for (int k = 0; k < KK; ++k) nbr[tid][k] = mi[k];
    }
    for (int i = tid; i < TILE * NB * 3; i += 256) (&G[0][0])[i] = 0.0f;
    __syncthreads();

    // ---- edge phase: thread = (row, k). rb[NB] x rhat[3] outer product ----
    {
        const int erow = tid >> 4;
        const int ek   = tid & 15;
        const int j    = nbr[erow][ek];
        const float* cj = &coords[((size_t)b * NN + j) * 3];
        float rx = cj[0] - rowC[erow][0];
        float ry = cj[1] - rowC[erow][1];
        float rz = cj[2] - rowC[erow][2];
        float dist = sqrtf(rx * rx + ry * ry + rz * rz);
        float inv  = 1.0f / (dist + 1e-8f);
        float hx = rx * inv, hy = ry * inv, hz = rz * inv;
        float cut = fminf(dist * (1.0f / MAX_RADIUS), 1.0f);
        float g[NB], s = 0.0f;
#pragma unroll
        for (int v = 0; v < NB; ++v) {
            float dc = cut - (float)v * (1.0f / 7.0f);   // linspace(0,1,8)
            g[v] = __expf(-32.0f * dc * dc);             // 1/(2*sigma^2), sigma=1/8
            s += g[v];
        }
        float invs = 1.0f / s;
#pragma unroll
        for (int v = 0; v < NB; ++v) {
            float rbv = g[v] * invs;
            atomicAdd(&G[erow][v * 3 + 0], rbv * hx);    // ds_add_f32
            atomicAdd(&G[erow][v * 3 + 1], rbv * hy);
            atomicAdd(&G[erow][v * 3 + 2], rbv * hz);
        }
    }
    __syncthreads();

    // ---- output: out[w,m] = (alpha/K) * sum_v W[v][w] * G[v][m] ----
    // 12 contiguous floats per thread -> 3 x global_store_b128.
    {
        const float scale = 0.35355339059327373f / (float)KK;  // 1/sqrt(NB)/K
        const int orow = tid >> 4;
        const int os   = tid & 15;
        float* dst = out + ((size_t)b * NN + rowBase + orow) * OUT_DIM + os * 12;
        float acc[12];
#pragma unroll
        for (int q = 0; q < 12; ++q) {
            const int oo = os * 12 + q;
            const int w = oo / 3, m = oo % 3;
            float a = 0.0f;
#pragma unroll
            for (int v = 0; v < NB; ++v)
                a += sW[v * OUT_MUL + w] * G[orow][v * 3 + m];
            acc[q] = a * scale;
        }
        float4* dst4 = reinterpret_cast<float4*>(dst);
#pragma unroll
        for (int q = 0; q < 3; ++q) {
            float4 vv;
            vv.x = acc[q * 4 + 0];
            vv.y = acc[q * 4 + 1];
            vv.z = acc[q * 4 + 2];
            vv.w = acc[q * 4 + 3];
            dst4[q] = vv;
        }
    }
}

extern "C" void kernel_launch(void* const* d_in, const int* in_sizes, int n_in,
                              void* d_out, int out_size, void* d_ws, size_t ws_size,
                              hipStream_t stream) {
    const float* coords = (const float*)d_in[0];   // [B,N,3] f32
    const float* tpW    = (const float*)d_in[1];   // [1,NB,OUT_MUL] f32
    float* out          = (float*)d_out;           // [B,N,192] f32
    float4* bcol        = (float4*)d_ws;           // B*N float4 = 512 KB scratch

    const int total = BB * NN;
    se3_prep_bcol<<<(total + 255) / 256, 256, 0, stream>>>(coords, bcol, total);

    dim3 grid(NN / TILE, BB);                      // 256 x 8 blocks
    se3_knn_tp_kernel<<<grid, 256, 0, stream>>>(coords, bcol, tpW, out);
}